// Model_7121055777102
// MI455X (gfx1250) — compile-verified
//
#include <hip/hip_runtime.h>

typedef float v2f __attribute__((ext_vector_type(2)));
typedef float v8f __attribute__((ext_vector_type(8)));
typedef unsigned int u32x4 __attribute__((ext_vector_type(4)));
typedef int i32x4 __attribute__((ext_vector_type(4)));
typedef int i32x8 __attribute__((ext_vector_type(8)));

constexpr int SEQ  = 2048;
constexpr int NB   = 64;     // batch
constexpr int ISZ  = 512;    // input size
constexpr int HSZ  = 512;    // hidden size
constexpr int MTOT = SEQ * NB;          // 131072 rows
constexpr int NBLK = 16;                // persistent blocks for the scan (512/32)

// Padded h layout in LDS produced by TDM (pad 2 dwords every 256 dwords):
// addr(row,k) = row*516 + k + 2*(k>=256)   [dwords], conflict-free A-frag reads
constexpr int HROW   = 516;                  // padded row stride in dwords
constexpr int BS_FP  = HSZ * 32;             // 16384 floats = 64KB (W_hh slice)
constexpr int HS_FP  = NB * HROW;            // 33024 floats = 129KB (padded h)
constexpr int SMEM_B = (BS_FP + HS_FP) * 4;  // 197632 bytes dynamic LDS

__device__ __forceinline__ v8f wmma_f32(v2f a, v2f b, v8f c) {
  // D = A(16x4 f32) * B(4x16 f32) + C(16x16 f32)
  return __builtin_amdgcn_wmma_f32_16x16x4_f32(
      false, a, false, b, (short)0, c, false, false);
}

// ---------------------------------------------------------------------------
// Phase 0: zero the h double-buffer and the grid-sync counter (runs every call)
// ---------------------------------------------------------------------------
__global__ void init_ws(float* __restrict__ hbuf, int* __restrict__ counter) {
  int i = blockIdx.x * blockDim.x + threadIdx.x;
  if (i < 2 * NB * HSZ) hbuf[i] = 0.0f;
  if (i == 0) *counter = 0;
}

// ---------------------------------------------------------------------------
// Phase 1: x_proj = inputs @ W_ih^T + (b_ih + b_hh), written into d_out.
// Grid: (MTOT/128, HSZ/64), 128 threads = 4 waves; each wave owns a 32x64 tile.
// ---------------------------------------------------------------------------
__global__ __launch_bounds__(128)
void xproj_kernel(const float* __restrict__ X,    // [MTOT, ISZ]
                  const float* __restrict__ Wih,  // [HSZ, ISZ]
                  const float* __restrict__ bih,  // [HSZ]
                  const float* __restrict__ bhh,  // [HSZ]
                  float* __restrict__ out)        // [MTOT, HSZ]
{
  const int lane   = threadIdx.x & 31;
  const int wave   = threadIdx.x >> 5;
  const int m_base = blockIdx.x * 128 + wave * 32;
  const int n_base = blockIdx.y * 64;
  const int lrow   = lane & 15;
  const int hi     = lane >> 4;        // 0 for lanes 0-15, 1 for 16-31
  const int khalf  = hi << 1;          // K sub-offset per the f32 A/B layout

  v8f acc[2][4];
  #pragma unroll
  for (int mt = 0; mt < 2; ++mt)
    #pragma unroll
    for (int nt = 0; nt < 4; ++nt)
      acc[mt][nt] = (v8f){0,0,0,0,0,0,0,0};

  for (int k = 0; k < ISZ; k += 4) {
    v2f a[2], b[4];
    #pragma unroll
    for (int mt = 0; mt < 2; ++mt)
      a[mt] = *(const v2f*)(X + (size_t)(m_base + mt * 16 + lrow) * ISZ + k + khalf);
    #pragma unroll
    for (int nt = 0; nt < 4; ++nt)
      b[nt] = *(const v2f*)(Wih + (size_t)(n_base + nt * 16 + lrow) * ISZ + k + khalf);
    #pragma unroll
    for (int mt = 0; mt < 2; ++mt)
      #pragma unroll
      for (int nt = 0; nt < 4; ++nt)
        acc[mt][nt] = wmma_f32(a[mt], b[nt], acc[mt][nt]);
  }

  #pragma unroll
  for (int nt = 0; nt < 4; ++nt) {
    const int col = n_base + nt * 16 + lrow;
    const float bias = bih[col] + bhh[col];
    #pragma unroll
    for (int mt = 0; mt < 2; ++mt)
      #pragma unroll
      for (int r = 0; r < 8; ++r) {
        const int row = m_base + mt * 16 + r + hi * 8;   // C layout: lanes 16-31 -> M+8
        out[(size_t)row * HSZ + col] = acc[mt][nt][r] + bias;
      }
  }
}

// ---------------------------------------------------------------------------
// Phase 2: persistent scan. 16 blocks x 256 threads (8 waves).
// Block b owns hidden columns [b*32, b*32+32); W_hh^T slice lives in LDS for
// the whole scan. h double-buffered in global (L2-resident). Per step:
//   - wave0 issues one TENSOR_LOAD_TO_LDS DMA of h_prev (128KB, TDM-padded
//     to a 516-dword row stride for bank-conflict-free A fragments)
//   - all waves load their xp fragment from d_out (overlaps the DMA)
//   - s_wait_tensorcnt + barrier, then K=512 of f32 WMMAs from LDS, split
//     over 4 independent accumulator chains to hide WMMA latency
//   - tanh, write h_t in place into d_out and into the other h buffer
//   - 16-block atomic grid barrier
// ---------------------------------------------------------------------------
__global__ __launch_bounds__(256)
void rnn_scan_kernel(float* __restrict__ io,        // d_out: xp in / h out
                     const float* __restrict__ Whh, // [HSZ, HSZ]
                     float* __restrict__ hbuf,      // 2 * NB * HSZ
                     int* __restrict__ counter)
{
  extern __shared__ float smem[];
  float* Bs = smem;            // [256][64]: (k/2)*64 + nl*2 + (k&1)
  float* Hs = smem + BS_FP;    // padded h tile written by TDM

  const int tid = threadIdx.x;
  const int blk = blockIdx.x;      // 0..15
  const int n0  = blk * 32;

  // one-time: W_hh^T slice -> LDS, K-pair interleaved for b64 B fragments
  for (int i = tid; i < HSZ * 32; i += 256) {
    const int nl = i >> 9;          // 0..31
    const int k  = i & (HSZ - 1);   // 0..511 (coalesced over k)
    Bs[(k >> 1) * 64 + nl * 2 + (k & 1)] = Whh[(size_t)(n0 + nl) * HSZ + k];
  }
  __syncthreads();

  // --- TDM descriptors (CDNA5 ISA) -----------------
  // 1D copy of 32768 f32 elements with pad: 2 dwords every 256 dwords.
  const unsigned lds_off = __builtin_amdgcn_groupstaticsize() + (unsigned)(BS_FP * 4);
  u32x4 g0d[2];
  #pragma unroll
  for (int p = 0; p < 2; ++p) {
    const unsigned long long ga =
        (unsigned long long)(uintptr_t)(hbuf + (size_t)p * NB * HSZ);
    g0d[p] = (u32x4){ 1u,                                  // count=1, user desc
                      lds_off,                             // lds_addr (bytes)
                      (unsigned)(ga & 0xFFFFFFFFu),        // global_addr[31:0]
                      (unsigned)((ga >> 32) & 0x01FFFFFFu) // global_addr[56:32]
                          | (2u << 30) };                  // type=2 ("image")
  }
  // group1: data_size=4B(2), pad_enable, pad_interval=256dw(7), pad_amount=2dw(1)
  //         tensor_dim0 = tile_dim0 = 32768
  const i32x8 g1d = (i32x8){ (int)0x03D20000,
                             (int)0x80000000,   // tensor_dim0[15:0]=0x8000 <<16
                             0,
                             (int)0x80000000,   // tile_dim0=0x8000 <<16
                             0, 0, 0, 0 };
  const i32x4 gz4 = (i32x4){0, 0, 0, 0};
  const i32x8 gz8 = (i32x8){0, 0, 0, 0, 0, 0, 0, 0};

  const int lane  = tid & 31;
  const int wave  = tid >> 5;      // 0..7
  const int mt    = wave >> 1;     // 0..3 -> M offset
  const int nt    = wave & 1;      // 0..1 -> N offset within the 32-col slice
  const int m0    = mt * 16;
  const int lrow  = lane & 15;
  const int hi    = lane >> 4;
  const int khalf = hi << 1;
  const int col_l = nt * 16 + lrow;    // 0..31 local column
  const int col_g = n0 + col_l;        // global hidden column

  const int arow_off = (m0 + lrow) * HROW;   // padded row for this lane's A rows

  for (int t = 0; t < SEQ; ++t) {
    float* __restrict__ hnext = hbuf + (size_t)((t + 1) & 1) * (NB * HSZ);
    float* __restrict__ xrow  = io + (size_t)t * NB * HSZ;

    // async DMA: h_prev (global, L2-hot) -> padded LDS tile
    if (wave == 0) {
      __builtin_amdgcn_tensor_load_to_lds(g0d[t & 1], g1d, gz4, gz4, gz8, 0);
    }

    // xp fragment (WMMA C layout) -- overlaps the tensor DMA; seeds chain 0
    v8f acc[4];
    #pragma unroll
    for (int r = 0; r < 8; ++r)
      acc[0][r] = xrow[(size_t)(m0 + r + hi * 8) * HSZ + col_g];
    acc[1] = (v8f){0,0,0,0,0,0,0,0};
    acc[2] = (v8f){0,0,0,0,0,0,0,0};
    acc[3] = (v8f){0,0,0,0,0,0,0,0};

    if (wave == 0) __builtin_amdgcn_s_wait_tensorcnt(0);
    __syncthreads();

    // h_prev @ Whh^T : 4 independent accumulator chains (k+4j -> acc[j]);
    // K split at 256 for the TDM pad (+2 dwords in the second half)
    const float* arow = Hs + arow_off;
    #pragma unroll 4
    for (int k = 0; k < 256; k += 16) {
      #pragma unroll
      for (int j = 0; j < 4; ++j) {
        const int kk = k + 4 * j + khalf;
        v2f a = *(const v2f*)(arow + kk);
        v2f b = *(const v2f*)(Bs + (kk >> 1) * 64 + col_l * 2);
        acc[j] = wmma_f32(a, b, acc[j]);
      }
    }
    #pragma unroll 4
    for (int k = 256; k < 512; k += 16) {
      #pragma unroll
      for (int j = 0; j < 4; ++j) {
        const int kk = k + 4 * j + khalf;
        v2f a = *(const v2f*)(arow + kk + 2);
        v2f b = *(const v2f*)(Bs + (kk >> 1) * 64 + col_l * 2);
        acc[j] = wmma_f32(a, b, acc[j]);
      }
    }
    const v8f accs = (acc[0] + acc[1]) + (acc[2] + acc[3]);

    // tanh + write h_t (double buffer + in place into d_out)
    #pragma unroll
    for (int r = 0; r < 8; ++r) {
      const float v = tanhf(accs[r]);
      const int row = m0 + r + hi * 8;
      hnext[(size_t)row * HSZ + col_g] = v;
      xrow[(size_t)row * HSZ + col_g]  = v;
    }

    // 16-block grid barrier (monotone counter; NBLK*(t+1) per generation)
    __syncthreads();
    if (tid == 0) {
      __hip_atomic_fetch_add(counter, 1, __ATOMIC_RELEASE, __HIP_MEMORY_SCOPE_AGENT);
      const int target = NBLK * (t + 1);
      while (__hip_atomic_load(counter, __ATOMIC_ACQUIRE, __HIP_MEMORY_SCOPE_AGENT) < target) {
        __builtin_amdgcn_s_sleep(1);
      }
    }
    __syncthreads();
  }
}

// ---------------------------------------------------------------------------
extern "C" void kernel_launch(void* const* d_in, const int* in_sizes, int n_in,
                              void* d_out, int out_size, void* d_ws, size_t ws_size,
                              hipStream_t stream) {
  const float* X   = (const float*)d_in[0];   // [SEQ, NB, ISZ]
  const float* Wih = (const float*)d_in[1];   // [HSZ, ISZ]
  const float* Whh = (const float*)d_in[2];   // [HSZ, HSZ]
  const float* bih = (const float*)d_in[3];   // [HSZ]
  const float* bhh = (const float*)d_in[4];   // [HSZ]
  float* out = (float*)d_out;                 // [SEQ*NB, HSZ]

  float* hbuf    = (float*)d_ws;                                  // 2*NB*HSZ floats
  int*   counter = (int*)((char*)d_ws + (size_t)2 * NB * HSZ * sizeof(float));

  // allow 193KB dynamic LDS (WGP has 320KB); capture-safe host call
  (void)hipFuncSetAttribute((const void*)rnn_scan_kernel,
                            hipFuncAttributeMaxDynamicSharedMemorySize, SMEM_B);

  init_ws<<<(2 * NB * HSZ + 255) / 256, 256, 0, stream>>>(hbuf, counter);

  dim3 g1(MTOT / 128, HSZ / 64);
  xproj_kernel<<<g1, 128, 0, stream>>>(X, Wih, bih, bhh, out);

  rnn_scan_kernel<<<NBLK, 256, SMEM_B, stream>>>(out, Whh, hbuf, counter);
}